// GraphConvolutionalNetwork_51891794870976
// MI455X (gfx1250) — compile-verified
//
#include <hip/hip_runtime.h>

typedef float v2f __attribute__((ext_vector_type(2)));
typedef float v8f __attribute__((ext_vector_type(8)));

#define DH   128
#define BN_EPS 1e-5f

// ---- hardware float atomic add (global_atomic_add_f32, no return) ----
__device__ __forceinline__ void gAtomicAddF(float* p, float v) {
  unsafeAtomicAdd(p, v);
}

// ---------------- degree / normalization ----------------
__global__ void k_deg_init(float* __restrict__ deg, int N) {
  int i = blockIdx.x * blockDim.x + threadIdx.x;
  if (i < N) deg[i] = 1.0f;                 // self-loop
}

__global__ void k_deg_acc(float* __restrict__ deg, const int* __restrict__ dst, int E) {
  int e = blockIdx.x * blockDim.x + threadIdx.x;
  if (e < E) gAtomicAddF(&deg[dst[e]], 1.0f);
}

__global__ void k_dinv(const float* __restrict__ deg, float* __restrict__ dinv, int N) {
  int i = blockIdx.x * blockDim.x + threadIdx.x;
  if (i < N) dinv[i] = rsqrtf(deg[i]);      // deg >= 1 always (self-loops)
}

// ---------------- fp32 WMMA GEMM: H[N,128] = X[N,128] @ W[128,128] ----------------
// Block = 256 threads = 8 waves; block owns 16 rows; wave w owns cols [16w,16w+16).
// K-loop of 128 in steps of 4 -> 32 x v_wmma_f32_16x16x4_f32 per wave.
__global__ __launch_bounds__(256) void k_gemm_xw(const float* __restrict__ X,
                                                 const float* __restrict__ W,
                                                 float* __restrict__ H, int N) {
  __shared__ float Xs[16][130];             // pad to avoid LDS bank conflicts
  const int row0 = blockIdx.x * 16;

  for (int t = threadIdx.x; t < 16 * DH; t += 256) {
    int r = t >> 7, c = t & 127;
    int rr = row0 + r; if (rr >= N) rr = N - 1;
    Xs[r][c] = X[(size_t)rr * DH + c];
  }
  __syncthreads();

  const int wave = threadIdx.x >> 5;        // col tile 0..7
  const int lane = threadIdx.x & 31;
  const int half = lane >> 4;               // 0: K pair {0,1}; 1: K pair {2,3}
  const int l16  = lane & 15;
  const int col0 = wave * 16;

  v8f acc = {};
#pragma unroll
  for (int k0 = 0; k0 < DH; k0 += 4) {
    v2f a, b;
    // A (16x4 f32): lane<16 -> M=l16,K={k0,k0+1}; lane>=16 -> K={k0+2,k0+3}
    a.x = Xs[l16][k0 + 2 * half + 0];
    a.y = Xs[l16][k0 + 2 * half + 1];
    // B (4x16 f32): rows striped across lanes, mirrored halves
    b.x = W[(size_t)(k0 + 2 * half + 0) * DH + col0 + l16];
    b.y = W[(size_t)(k0 + 2 * half + 1) * DH + col0 + l16];
    acc = __builtin_amdgcn_wmma_f32_16x16x4_f32(false, a, false, b,
                                                (short)0, acc, false, false);
  }

  // C/D layout: VGPR r -> M = r + 8*half, N = l16
#pragma unroll
  for (int r = 0; r < 8; ++r) {
    int row = row0 + r + half * 8;
    if (row < N) H[(size_t)row * DH + col0 + l16] = acc[r];
  }
}

// ---------------- aggregation ----------------
// AGG[i][c] = bias[c] + H[i][c]*dinv[i]^2   (self-loop + bias folded in)
__global__ void k_agg_init(const float* __restrict__ H, const float* __restrict__ dinv,
                           const float* __restrict__ bias, float* __restrict__ AGG,
                           int total) {
  int idx = blockIdx.x * blockDim.x + threadIdx.x;
  if (idx >= total) return;
  int i = idx >> 7, c = idx & 127;
  float di = dinv[i];
  AGG[idx] = bias[c] + H[idx] * di * di;
}

// one wave per edge: float4 gather of H[src], 4 float atomics into AGG[dst]
__global__ __launch_bounds__(256) void k_edge_scatter(const float* __restrict__ H,
                                                      const int* __restrict__ src,
                                                      const int* __restrict__ dst,
                                                      const float* __restrict__ dinv,
                                                      float* __restrict__ AGG, int E) {
  int gw   = (blockIdx.x * 256 + threadIdx.x) >> 5;
  int lane = threadIdx.x & 31;
  if (gw >= E) return;
  int s = src[gw], d = dst[gw];
  float w = dinv[s] * dinv[d];
  float4 h = ((const float4*)(H + (size_t)s * DH))[lane];
  float* p = AGG + (size_t)d * DH + lane * 4;
  gAtomicAddF(p + 0, h.x * w);
  gAtomicAddF(p + 1, h.y * w);
  gAtomicAddF(p + 2, h.z * w);
  gAtomicAddF(p + 3, h.w * w);
}

// ---------------- batch norm ----------------
__global__ void k_zero_stats(float* __restrict__ s) {   // zeroes gsum[128]+gsq[128]
  if (threadIdx.x < 256) s[threadIdx.x] = 0.0f;
}

__global__ __launch_bounds__(128) void k_bn_stats(const float* __restrict__ A,
                                                  float* __restrict__ gsum,
                                                  float* __restrict__ gsq, int N) {
  int c = threadIdx.x;                       // channel
  float s = 0.f, q = 0.f;
  for (int r = blockIdx.x; r < N; r += gridDim.x) {
    float v = A[(size_t)r * DH + c];
    s += v; q += v * v;
  }
  gAtomicAddF(&gsum[c], s);
  gAtomicAddF(&gsq[c], q);
}

__global__ __launch_bounds__(128) void k_bn_final(const float* __restrict__ gsum,
                                                  const float* __restrict__ gsq,
                                                  const float* __restrict__ g,
                                                  const float* __restrict__ be,
                                                  float* __restrict__ scale,
                                                  float* __restrict__ shift, int N) {
  int c = threadIdx.x;
  float inv  = 1.0f / (float)N;
  float mean = gsum[c] * inv;
  float var  = gsq[c] * inv - mean * mean;
  float a    = g[c] * rsqrtf(var + BN_EPS);
  scale[c] = a;
  shift[c] = be[c] - mean * a;
}

__global__ void k_bn_apply(const float* __restrict__ A, const float* __restrict__ scale,
                           const float* __restrict__ shift, float* __restrict__ O,
                           int total, int relu) {
  int idx = blockIdx.x * blockDim.x + threadIdx.x;
  if (idx >= total) return;
  int c = idx & 127;
  float v = A[idx] * scale[c] + shift[c];
  if (relu) v = fmaxf(v, 0.0f);
  O[idx] = v;
}

// ---------------- classifier: out[N,2] = emb @ Wc + bc ----------------
__global__ __launch_bounds__(256) void k_classifier(const float* __restrict__ emb,
                                                    const float* __restrict__ Wc,
                                                    const float* __restrict__ bc,
                                                    float* __restrict__ out, int N) {
  int gw   = (blockIdx.x * 256 + threadIdx.x) >> 5;
  int lane = threadIdx.x & 31;
  if (gw >= N) return;
  float4 e = ((const float4*)(emb + (size_t)gw * DH))[lane];
  int c = lane * 4;
  float a0 = e.x * Wc[(c + 0) * 2 + 0] + e.y * Wc[(c + 1) * 2 + 0] +
             e.z * Wc[(c + 2) * 2 + 0] + e.w * Wc[(c + 3) * 2 + 0];
  float a1 = e.x * Wc[(c + 0) * 2 + 1] + e.y * Wc[(c + 1) * 2 + 1] +
             e.z * Wc[(c + 2) * 2 + 1] + e.w * Wc[(c + 3) * 2 + 1];
#pragma unroll
  for (int off = 16; off > 0; off >>= 1) {
    a0 += __shfl_down(a0, off, 32);
    a1 += __shfl_down(a1, off, 32);
  }
  if (lane == 0) {
    out[(size_t)gw * 2 + 0] = a0 + bc[0];
    out[(size_t)gw * 2 + 1] = a1 + bc[1];
  }
}

// ---------------- launcher ----------------
extern "C" void kernel_launch(void* const* d_in, const int* in_sizes, int n_in,
                              void* d_out, int out_size, void* d_ws, size_t ws_size,
                              hipStream_t stream) {
  const float* x  = (const float*)d_in[0];
  const int*   ei = (const int*)d_in[1];
  const float* W1 = (const float*)d_in[2];  const float* b1 = (const float*)d_in[3];
  const float* g1 = (const float*)d_in[4];  const float* be1 = (const float*)d_in[5];
  const float* W2 = (const float*)d_in[6];  const float* b2 = (const float*)d_in[7];
  const float* g2 = (const float*)d_in[8];  const float* be2 = (const float*)d_in[9];
  const float* W3 = (const float*)d_in[10]; const float* b3 = (const float*)d_in[11];
  const float* g3 = (const float*)d_in[12]; const float* be3 = (const float*)d_in[13];
  const float* Wc = (const float*)d_in[14]; const float* bc = (const float*)d_in[15];

  const int N = in_sizes[0] / DH;
  const int E = in_sizes[1] / 2;
  const int* src = ei;
  const int* dst = ei + E;

  float* ws   = (float*)d_ws;
  float* deg  = ws;                              // N
  float* dinv = ws + (size_t)N;                  // N
  float* B0   = ws + (size_t)2 * N;              // N*128
  float* B1   = B0 + (size_t)N * DH;             // N*128
  float* gsum = B1 + (size_t)N * DH;             // 128
  float* gsq  = gsum + 128;                      // 128
  float* scl  = gsq + 128;                       // 128
  float* shf  = scl + 128;                       // 128

  float* out = (float*)d_out;                    // N*2
  float* emb = out + (size_t)N * 2;              // N*128

  const int total = N * DH;
  const dim3 B(256);
  const int gN  = (N + 255) / 256;
  const int gE  = (E + 255) / 256;
  const int gT  = (total + 255) / 256;
  const int gEw = (E + 7) / 8;                   // one wave per edge
  const int gM  = (N + 15) / 16;                 // GEMM row tiles

  // normalization coefficients
  k_deg_init<<<gN, B, 0, stream>>>(deg, N);
  k_deg_acc <<<gE, B, 0, stream>>>(deg, dst, E);
  k_dinv    <<<gN, B, 0, stream>>>(deg, dinv, N);

  // ---- layer 1: x -> B0(H) -> B1(AGG) -> B0(relu(bn)) ----
  k_gemm_xw     <<<gM, B, 0, stream>>>(x, W1, B0, N);
  k_agg_init    <<<gT, B, 0, stream>>>(B0, dinv, b1, B1, total);
  k_edge_scatter<<<gEw, B, 0, stream>>>(B0, src, dst, dinv, B1, E);
  k_zero_stats  <<<1, B, 0, stream>>>(gsum);
  k_bn_stats    <<<512, 128, 0, stream>>>(B1, gsum, gsq, N);
  k_bn_final    <<<1, 128, 0, stream>>>(gsum, gsq, g1, be1, scl, shf, N);
  k_bn_apply    <<<gT, B, 0, stream>>>(B1, scl, shf, B0, total, 1);

  // ---- layer 2: B0 -> B1(H) -> B0(AGG) -> B1(relu(bn)) ----
  k_gemm_xw     <<<gM, B, 0, stream>>>(B0, W2, B1, N);
  k_agg_init    <<<gT, B, 0, stream>>>(B1, dinv, b2, B0, total);
  k_edge_scatter<<<gEw, B, 0, stream>>>(B1, src, dst, dinv, B0, E);
  k_zero_stats  <<<1, B, 0, stream>>>(gsum);
  k_bn_stats    <<<512, 128, 0, stream>>>(B0, gsum, gsq, N);
  k_bn_final    <<<1, 128, 0, stream>>>(gsum, gsq, g2, be2, scl, shf, N);
  k_bn_apply    <<<gT, B, 0, stream>>>(B0, scl, shf, B1, total, 1);

  // ---- layer 3: B1 -> B0(H) -> B1(AGG) -> emb(bn, no relu) ----
  k_gemm_xw     <<<gM, B, 0, stream>>>(B1, W3, B0, N);
  k_agg_init    <<<gT, B, 0, stream>>>(B0, dinv, b3, B1, total);
  k_edge_scatter<<<gEw, B, 0, stream>>>(B0, src, dst, dinv, B1, E);
  k_zero_stats  <<<1, B, 0, stream>>>(gsum);
  k_bn_stats    <<<512, 128, 0, stream>>>(B1, gsum, gsq, N);
  k_bn_final    <<<1, 128, 0, stream>>>(gsum, gsq, g3, be3, scl, shf, N);
  k_bn_apply    <<<gT, B, 0, stream>>>(B1, scl, shf, emb, total, 0);

  // ---- classifier ----
  k_classifier<<<(N + 7) / 8, B, 0, stream>>>(emb, Wc, bc, out, N);
}